// GATv2model_70351564308947
// MI455X (gfx1250) — compile-verified
//
#include <hip/hip_runtime.h>
#include <hip/hip_bf16.h>

#define HEADS 3
#define FHID 64
#define HC 192            // HEADS * FHID
#define NEG_SLOPE 0.2f

typedef __attribute__((ext_vector_type(2))) float v2f;
typedef __attribute__((ext_vector_type(8))) float v8f;

// ---------------------------------------------------------------------------
// fp32 WMMA GEMM:  out[M x HC] = A[M x K] * W[K x HC] + bias   (row-major)
// One wave32 computes one 16x16 output tile via V_WMMA_F32_16X16X4_F32.
// M % 16 == 0 (50000 = 3125*16), HC = 192 = 12*16, K in {128, 64}.
// Wave-uniform guard keeps EXEC all-1s for WMMA.
// ---------------------------------------------------------------------------
__global__ __launch_bounds__(256)
void gemm_wmma_f32(const float* __restrict__ A, const float* __restrict__ W,
                   const float* __restrict__ bias, float* __restrict__ out,
                   int M, int K) {
  const int tilesN = HC / 16;                     // 12
  const int totalTiles = (M / 16) * tilesN;
  int wid = blockIdx.x * (blockDim.x >> 5) + (threadIdx.x >> 5);
  if (wid >= totalTiles) return;                  // whole wave exits together
  int tm = wid / tilesN, tn = wid - tm * tilesN;
  int lane   = threadIdx.x & 31;
  int laneLo = lane & 15;                         // M (A) / N (B,C)
  int laneHi = lane >> 4;                         // selects K+2 half / M+8 half

  // A fragment source: row tm*16+laneLo, cols k0+2*laneHi .. +1  (8B aligned)
  const float* arow = A + (size_t)(tm * 16 + laneLo) * K + 2 * laneHi;
  // B fragment source: col tn*16+laneLo, rows k0+2*laneHi .. +1
  const float* wcol = W + (tn * 16 + laneLo);

  v8f acc = {};
  for (int k0 = 0; k0 < K; k0 += 4) {
    float2 av = *reinterpret_cast<const float2*>(arow + k0);
    v2f a; a[0] = av.x; a[1] = av.y;
    const float* wp = wcol + (size_t)(k0 + 2 * laneHi) * HC;
    v2f b; b[0] = wp[0]; b[1] = wp[HC];
    // 8 args: (neg_a, A, neg_b, B, c_mod, C, reuse_a, reuse_b)
    acc = __builtin_amdgcn_wmma_f32_16x16x4_f32(
        false, a, false, b, (short)0, acc, false, false);
  }

  float bv = bias[tn * 16 + laneLo];
  // C/D layout: VGPR j, element (M = j + 8*laneHi, N = laneLo)
  float* orow = out + (size_t)(tm * 16 + 8 * laneHi) * HC + tn * 16 + laneLo;
#pragma unroll
  for (int j = 0; j < 8; ++j)
    orow[(size_t)j * HC] = acc[j] + bv;
}

// --- monotonic uint encoding of float for atomicMax-based segment max ------
__device__ __forceinline__ unsigned fenc(float f) {
  unsigned u = __float_as_uint(f);
  return (u & 0x80000000u) ? ~u : (u | 0x80000000u);
}
__device__ __forceinline__ float fdec(unsigned u) {
  u = (u & 0x80000000u) ? (u & 0x7FFFFFFFu) : ~u;
  return __uint_as_float(u);
}

// ---------------------------------------------------------------------------
// Per-edge attention logits + segment max.  One wave per edge; lane handles
// channels (lane, lane+32) per head; wave shuffle reduction (wave32).
// ---------------------------------------------------------------------------
__global__ __launch_bounds__(256)
void edge_logits(const float* __restrict__ xl, const float* __restrict__ xr,
                 const float* __restrict__ att, const int* __restrict__ ei,
                 float* __restrict__ elog, unsigned* __restrict__ menc,
                 int NE, int ET) {
  int e = blockIdx.x * (blockDim.x >> 5) + (threadIdx.x >> 5);
  if (e >= ET) return;
  int lane = threadIdx.x & 31;
  int src, dst;
  if (e < NE) { src = ei[e]; dst = ei[NE + e]; }
  else        { src = e - NE; dst = src; }              // appended self-loops
  const float* pl = xl + (size_t)src * HC;
  const float* pr = xr + (size_t)dst * HC;
#pragma unroll
  for (int h = 0; h < HEADS; ++h) {
    float g0 = pl[h * 64 + lane]      + pr[h * 64 + lane];
    float g1 = pl[h * 64 + lane + 32] + pr[h * 64 + lane + 32];
    g0 = g0 > 0.f ? g0 : NEG_SLOPE * g0;
    g1 = g1 > 0.f ? g1 : NEG_SLOPE * g1;
    float part = g0 * att[h * 64 + lane] + g1 * att[h * 64 + lane + 32];
#pragma unroll
    for (int off = 16; off > 0; off >>= 1)
      part += __shfl_xor(part, off, 32);
    if (lane == 0) {
      elog[(size_t)e * HEADS + h] = part;
      atomicMax(&menc[dst * HEADS + h], fenc(part));
    }
  }
}

// ---------------------------------------------------------------------------
// p = exp(e - max[dst]);  segment sum via atomicAdd.  Thread per (edge, head).
// ---------------------------------------------------------------------------
__global__ __launch_bounds__(256)
void edge_softmax_num(const int* __restrict__ ei, float* __restrict__ elog,
                      const unsigned* __restrict__ menc,
                      float* __restrict__ ssum, int NE, int ET) {
  int idx = blockIdx.x * blockDim.x + threadIdx.x;
  if (idx >= ET * HEADS) return;
  int e = idx / HEADS, h = idx - e * HEADS;
  int dst = (e < NE) ? ei[NE + e] : (e - NE);
  float m = fdec(menc[dst * HEADS + h]);
  float p = __expf(elog[idx] - m);
  elog[idx] = p;
  atomicAdd(&ssum[dst * HEADS + h], p);
}

// ---------------------------------------------------------------------------
// accum[dst] += xl[src] * (p / (s[dst]+eps))  — wave per edge.
// ---------------------------------------------------------------------------
__global__ __launch_bounds__(256)
void edge_aggregate(const float* __restrict__ xl, const int* __restrict__ ei,
                    const float* __restrict__ elog, const float* __restrict__ ssum,
                    float* __restrict__ accum, int NE, int ET) {
  int e = blockIdx.x * (blockDim.x >> 5) + (threadIdx.x >> 5);
  if (e >= ET) return;
  int lane = threadIdx.x & 31;
  int src, dst;
  if (e < NE) { src = ei[e]; dst = ei[NE + e]; }
  else        { src = e - NE; dst = src; }
  const float* pl = xl + (size_t)src * HC;
  float* pa = accum + (size_t)dst * HC;
#pragma unroll
  for (int h = 0; h < HEADS; ++h) {
    float a = elog[(size_t)e * HEADS + h] / (ssum[dst * HEADS + h] + 1e-16f);
    atomicAdd(&pa[h * 64 + lane],      pl[h * 64 + lane]      * a);
    atomicAdd(&pa[h * 64 + lane + 32], pl[h * 64 + lane + 32] * a);
  }
}

// ---------------------------------------------------------------------------
// Head mean + bias (+ optional ReLU).  Thread per (node, channel).
// ---------------------------------------------------------------------------
__global__ __launch_bounds__(256)
void finalize_mean(const float* __restrict__ accum, const float* __restrict__ bias,
                   float* __restrict__ out, int N, int applyRelu) {
  int idx = blockIdx.x * blockDim.x + threadIdx.x;
  if (idx >= N * FHID) return;
  int n = idx / FHID, c = idx - n * FHID;
  const float* pa = accum + (size_t)n * HC;
  float v = (pa[c] + pa[64 + c] + pa[128 + c]) * (1.0f / 3.0f) + bias[c];
  if (applyRelu) v = v > 0.f ? v : 0.f;
  out[idx] = v;
}

extern "C" void kernel_launch(void* const* d_in, const int* in_sizes, int n_in,
                              void* d_out, int out_size, void* d_ws, size_t ws_size,
                              hipStream_t stream) {
  const float* x    = (const float*)d_in[0];
  const int*   ei   = (const int*)d_in[1];
  const float* W1l  = (const float*)d_in[2];
  const float* b1l  = (const float*)d_in[3];
  const float* W1r  = (const float*)d_in[4];
  const float* b1r  = (const float*)d_in[5];
  const float* att1 = (const float*)d_in[6];
  const float* bias1= (const float*)d_in[7];
  const float* W2l  = (const float*)d_in[8];
  const float* b2l  = (const float*)d_in[9];
  const float* W2r  = (const float*)d_in[10];
  const float* b2r  = (const float*)d_in[11];
  const float* att2 = (const float*)d_in[12];
  const float* bias2= (const float*)d_in[13];
  float* out = (float*)d_out;

  const int FIN = 128;
  const int N  = in_sizes[0] / FIN;     // 50000
  const int NE = in_sizes[1] / 2;       // 800000
  const int ET = NE + N;                // + self loops

  // ---- workspace carve-out (floats) ----
  float* ws = (float*)d_ws;
  size_t o = 0;
  float*    xl    = ws + o; o += (size_t)N * HC;
  float*    xr    = ws + o; o += (size_t)N * HC;
  float*    accum = ws + o; o += (size_t)N * HC;
  float*    hbuf  = ws + o; o += (size_t)N * FHID;
  float*    elog  = ws + o; o += (size_t)ET * HEADS;
  unsigned* menc  = (unsigned*)(ws + o); o += (size_t)N * HEADS;
  float*    ssum  = ws + o; o += (size_t)N * HEADS;

  const int tiles  = (N / 16) * (HC / 16);
  const int gGemm  = (tiles + 7) / 8;            // 8 waves / block
  const int gEdgeW = (ET + 7) / 8;               // wave per edge
  const int gEdgeT = (ET * HEADS + 255) / 256;
  const int gFin   = (N * FHID + 255) / 256;
  dim3 blk(256);

  // =================== layer 1 (K = 128) ===================
  gemm_wmma_f32<<<gGemm, blk, 0, stream>>>(x, W1l, b1l, xl, N, FIN);
  gemm_wmma_f32<<<gGemm, blk, 0, stream>>>(x, W1r, b1r, xr, N, FIN);
  hipMemsetAsync(menc,  0, (size_t)N * HEADS * sizeof(unsigned), stream);
  hipMemsetAsync(ssum,  0, (size_t)N * HEADS * sizeof(float),    stream);
  hipMemsetAsync(accum, 0, (size_t)N * HC    * sizeof(float),    stream);
  edge_logits     <<<gEdgeW, blk, 0, stream>>>(xl, xr, att1, ei, elog, menc, NE, ET);
  edge_softmax_num<<<gEdgeT, blk, 0, stream>>>(ei, elog, menc, ssum, NE, ET);
  edge_aggregate  <<<gEdgeW, blk, 0, stream>>>(xl, ei, elog, ssum, accum, NE, ET);
  finalize_mean   <<<gFin,   blk, 0, stream>>>(accum, bias1, hbuf, N, /*relu=*/1);

  // =================== layer 2 (K = 64) ====================
  gemm_wmma_f32<<<gGemm, blk, 0, stream>>>(hbuf, W2l, b2l, xl, N, FHID);
  gemm_wmma_f32<<<gGemm, blk, 0, stream>>>(hbuf, W2r, b2r, xr, N, FHID);
  hipMemsetAsync(menc,  0, (size_t)N * HEADS * sizeof(unsigned), stream);
  hipMemsetAsync(ssum,  0, (size_t)N * HEADS * sizeof(float),    stream);
  hipMemsetAsync(accum, 0, (size_t)N * HC    * sizeof(float),    stream);
  edge_logits     <<<gEdgeW, blk, 0, stream>>>(xl, xr, att2, ei, elog, menc, NE, ET);
  edge_softmax_num<<<gEdgeT, blk, 0, stream>>>(ei, elog, menc, ssum, NE, ET);
  edge_aggregate  <<<gEdgeW, blk, 0, stream>>>(xl, ei, elog, ssum, accum, NE, ET);
  finalize_mean   <<<gFin,   blk, 0, stream>>>(accum, bias2, out, N, /*relu=*/0);
}